// GNN_Layer_57217554317352
// MI455X (gfx1250) — compile-verified
//
#include <hip/hip_runtime.h>

typedef __attribute__((ext_vector_type(2))) float v2f;
typedef __attribute__((ext_vector_type(8))) float v8f;

#define F 128            // F_IN == F_OUT == 128
#define TILES_PER_BLOCK 8

// -----------------------------------------------------------------------------
// Kernel 1: support = X @ W using V_WMMA_F32_16X16X4_F32 (full f32 precision).
// One wave computes a 16-row strip across all 128 output columns:
//   8 accumulators (16x16 tiles), A fragment reused 8x per K-step.
// A 16x4 f32 layout (ISA 7.12.2): lane = 16*h + r (h=0/1, r=0..15),
//   VGPR j holds element (M=r, K=2*h+j).  B (4x16) mirrors: VGPR j holds
//   (K=2*h+j, N=r).  C/D 16x16 f32: VGPR v, lane 16*h+r -> (M=v+8*h, N=r).
// -----------------------------------------------------------------------------
__global__ void gcn_gemm_wmma(const float* __restrict__ x,
                              const float* __restrict__ W,
                              float* __restrict__ support,
                              int N) {
  const int lane = threadIdx.x & 31;
  const int wave = threadIdx.x >> 5;
  const int tile = blockIdx.x * TILES_PER_BLOCK + wave;   // 16-row tile index
  if (tile * 16 >= N) return;                             // wave-uniform guard

  const int r = lane & 15;   // 0..15
  const int h = lane >> 4;   // 0..1

  const int rowA = tile * 16 + r;
  const int rowAc = rowA < N ? rowA : N - 1;              // clamp (N%16==0 here)
  const float* __restrict__ xrow = x + (size_t)rowAc * F;

  v8f acc[8];
  const v8f zero = {0.f, 0.f, 0.f, 0.f, 0.f, 0.f, 0.f, 0.f};
#pragma unroll
  for (int c = 0; c < 8; ++c) acc[c] = zero;

  for (int k = 0; k < F; k += 4) {
    // A fragment: two K-contiguous floats -> one 8-byte load
    const v2f a = *reinterpret_cast<const v2f*>(xrow + k + 2 * h);
    // B fragments for the 8 column tiles (W is K-major [128x128], cache-hot)
#pragma unroll
    for (int c = 0; c < 8; ++c) {
      v2f b;
      b.x = W[(size_t)(k + 2 * h + 0) * F + c * 16 + r];
      b.y = W[(size_t)(k + 2 * h + 1) * F + c * 16 + r];
      acc[c] = __builtin_amdgcn_wmma_f32_16x16x4_f32(
          /*neg_a=*/false, a, /*neg_b=*/false, b,
          /*c_mod=*/(short)0, acc[c], /*reuse_a=*/false, /*reuse_b=*/false);
    }
  }

  // Store D: lanes 0..15 write consecutive columns -> coalesced 64B per half
#pragma unroll
  for (int v = 0; v < 8; ++v) {
    const int row = tile * 16 + v + 8 * h;
    if (row < N) {
#pragma unroll
      for (int c = 0; c < 8; ++c) {
        support[(size_t)row * F + c * 16 + r] = acc[c][v];
      }
    }
  }
}

// -----------------------------------------------------------------------------
// Kernel 2: out[i, f] = b[f]  (bias folded into the accumulation base; also
// deterministically re-initializes d_out each call).
// -----------------------------------------------------------------------------
__global__ void gcn_init_out(float* __restrict__ out,
                             const float* __restrict__ bias,
                             int total) {
  const int idx = blockIdx.x * blockDim.x + threadIdx.x;
  if (idx < total) out[idx] = bias[idx & (F - 1)];
}

// -----------------------------------------------------------------------------
// Kernel 3: edge scatter.  One wave per edge; metadata is wave-uniform
// (scalar loads), each lane owns 4 contiguous floats (float4 gather +
// 4 relaxed agent-scope f32 atomic adds -> global_atomic_add_f32, L2-resident).
// -----------------------------------------------------------------------------
__global__ void gcn_scatter(const int* __restrict__ erow,
                            const int* __restrict__ ecol,
                            const float* __restrict__ eval,
                            const float* __restrict__ support,
                            float* __restrict__ out,
                            int E) {
  const int lane = threadIdx.x & 31;
  const int e = blockIdx.x * (blockDim.x >> 5) + (threadIdx.x >> 5);
  if (e >= E) return;

  const int dst = erow[e];
  const int src = ecol[e];
  const float v = eval[e];

  const float4 s =
      *reinterpret_cast<const float4*>(support + (size_t)src * F + lane * 4);
  float* o = out + (size_t)dst * F + lane * 4;

  __hip_atomic_fetch_add(o + 0, v * s.x, __ATOMIC_RELAXED, __HIP_MEMORY_SCOPE_AGENT);
  __hip_atomic_fetch_add(o + 1, v * s.y, __ATOMIC_RELAXED, __HIP_MEMORY_SCOPE_AGENT);
  __hip_atomic_fetch_add(o + 2, v * s.z, __ATOMIC_RELAXED, __HIP_MEMORY_SCOPE_AGENT);
  __hip_atomic_fetch_add(o + 3, v * s.w, __ATOMIC_RELAXED, __HIP_MEMORY_SCOPE_AGENT);
}

// -----------------------------------------------------------------------------
// Launch: GEMM -> init(out=b) -> scatter, all on `stream` (stream-ordered).
// d_ws holds support [N x 128] f32 (51.2 MB).
// -----------------------------------------------------------------------------
extern "C" void kernel_launch(void* const* d_in, const int* in_sizes, int n_in,
                              void* d_out, int out_size, void* d_ws, size_t ws_size,
                              hipStream_t stream) {
  const float* x    = (const float*)d_in[0];
  const int*   erow = (const int*)d_in[1];
  const int*   ecol = (const int*)d_in[2];
  const float* eval = (const float*)d_in[3];
  const float* W    = (const float*)d_in[4];
  const float* bias = (const float*)d_in[5];

  float* out     = (float*)d_out;
  float* support = (float*)d_ws;

  const int N = in_sizes[0] / F;   // 100000 nodes
  const int E = in_sizes[1];       // 3200000 edges

  const int tiles = (N + 15) / 16;
  const int gemmBlocks = (tiles + TILES_PER_BLOCK - 1) / TILES_PER_BLOCK;
  gcn_gemm_wmma<<<gemmBlocks, 32 * TILES_PER_BLOCK, 0, stream>>>(x, W, support, N);

  const int total = N * F;
  gcn_init_out<<<(total + 255) / 256, 256, 0, stream>>>(out, bias, total);

  const int edgesPerBlock = 256 / 32;
  const int scatBlocks = (E + edgesPerBlock - 1) / edgesPerBlock;
  gcn_scatter<<<scatBlocks, 256, 0, stream>>>(erow, ecol, eval, support, out, E);
}